// DummyGAT_75067438399515
// MI455X (gfx1250) — compile-verified
//
#include <hip/hip_runtime.h>

typedef __attribute__((ext_vector_type(2))) float v2f;
typedef __attribute__((ext_vector_type(8))) float v8f;

#define NEG_SLOPE 0.2f
#define HID 64

// ---------- order-preserving float <-> uint encoding for atomic max ----------
__device__ __forceinline__ unsigned int f32_enc(float f) {
    unsigned int b = __float_as_uint(f);
    return (b & 0x80000000u) ? ~b : (b | 0x80000000u);
}
__device__ __forceinline__ float f32_dec(unsigned int u) {
    unsigned int b = (u & 0x80000000u) ? (u ^ 0x80000000u) : ~u;
    return __uint_as_float(b);
}

// ---------- init: zero agg / denom / sumvec, menc = encoding floor ----------
__global__ void gat_init(float* agg, unsigned int* menc, float* denom,
                         float* sumvec, long long N) {
    long long t = (long long)blockIdx.x * blockDim.x + threadIdx.x;
    long long tot = N * HID;
    if (t < tot) agg[t] = 0.0f;
    if (t < N) { menc[t] = 0u; denom[t] = 0.0f; }
    if (t < HID) sumvec[t] = 0.0f;
}

// ---------- h = x @ W via V_WMMA_F32_16X16X4_F32 ----------
// block = 256 threads = 8 waves; block covers 32 rows x 64 cols,
// each wave owns one 16x16 tile; K=128 -> 32 wmma steps.
__global__ __launch_bounds__(256) void gat_gemm_h(const float* __restrict__ x,
                                                  const float* __restrict__ W,
                                                  float* __restrict__ h,
                                                  long long N) {
    const int lane   = threadIdx.x & 31;
    const int waveId = threadIdx.x >> 5;
    const int rowT   = (waveId >> 2) * 16;       // 0 or 16
    const int colT   = (waveId & 3) * 16;        // 0,16,32,48
    long long row0   = (long long)blockIdx.x * 32 + rowT;
    if (row0 >= N) return;

    const int lane15 = lane & 15;
    const int kOff   = (lane >> 4) * 2;          // 0 for lanes 0-15, 2 for 16-31
    long long aRow   = row0 + lane15;
    if (aRow > N - 1) aRow = N - 1;              // clamp loads (stores guarded)

    const float* xrow = x + aRow * 128;
    v8f c = {};
    #pragma unroll 4
    for (int k0 = 0; k0 < 128; k0 += 4) {
        // A fragment: lane<16 -> A[M=lane][k0..k0+1]; lane>=16 -> K=k0+2..k0+3
        v2f a = *(const v2f*)(xrow + k0 + kOff);
        // B fragment: VGPR0 = row (k0+kOff), VGPR1 = row (k0+kOff+1), N = lane&15
        v2f b;
        b.x = W[(k0 + kOff    ) * HID + colT + lane15];
        b.y = W[(k0 + kOff + 1) * HID + colT + lane15];
        c = __builtin_amdgcn_wmma_f32_16x16x4_f32(false, a, false, b,
                                                  (short)0, c, false, false);
    }
    // C/D layout: VGPR i -> M = i (lanes 0-15) or M = i+8 (lanes 16-31), N = lane&15
    const int mBase = (lane >> 4) * 8;
    float* hp = h + (row0 + mBase) * HID + colT + lane15;
    if (row0 + 16 <= N) {
        // full tile (always taken when N % 32 == 0): one clause of 8 offset stores
        #pragma unroll
        for (int i = 0; i < 8; ++i) hp[i * HID] = c[i];
    } else {
        #pragma unroll
        for (int i = 0; i < 8; ++i)
            if (row0 + mBase + i < N) hp[i * HID] = c[i];
    }
}

// ---------- per-node attention scores a_src, a_dst ----------
__global__ void gat_scores(const float* __restrict__ h,
                           const float* __restrict__ att_src,
                           const float* __restrict__ att_dst,
                           float* __restrict__ a_src, float* __restrict__ a_dst,
                           long long N) {
    long long n = (long long)blockIdx.x * blockDim.x + threadIdx.x;
    if (n >= N) return;
    const float* hr = h + n * HID;
    float s = 0.0f, d = 0.0f;
    #pragma unroll
    for (int f = 0; f < HID; f += 4) {
        float4 hv = *(const float4*)(hr + f);
        float4 as = *(const float4*)(att_src + f);
        float4 ad = *(const float4*)(att_dst + f);
        s += hv.x * as.x + hv.y * as.y + hv.z * as.z + hv.w * as.w;
        d += hv.x * ad.x + hv.y * ad.y + hv.z * ad.z + hv.w * ad.w;
    }
    a_src[n] = s;
    a_dst[n] = d;
}

// ---------- pass 1: leaky-relu logits + segment max (encoded atomicMax) ----------
__global__ void gat_edge_max(const int* __restrict__ src, const int* __restrict__ dst,
                             const float* __restrict__ a_src, const float* __restrict__ a_dst,
                             float* __restrict__ logits, unsigned int* __restrict__ menc,
                             long long E, long long ET) {
    long long t = (long long)blockIdx.x * blockDim.x + threadIdx.x;
    if (t >= ET) return;
    int s, d;
    if (t < E) { s = src[t]; d = dst[t]; }
    else       { s = d = (int)(t - E); }            // self loop
    float l = a_src[s] + a_dst[d];
    l = (l > 0.0f) ? l : NEG_SLOPE * l;
    logits[t] = l;
    atomicMax(&menc[d], f32_enc(l));
}

// ---------- pass 2: exp(logit - max) + segment sum ----------
__global__ void gat_edge_expsum(const int* __restrict__ dst,
                                float* __restrict__ logits,
                                const unsigned int* __restrict__ menc,
                                float* __restrict__ denom,
                                long long E, long long ET) {
    long long t = (long long)blockIdx.x * blockDim.x + threadIdx.x;
    if (t >= ET) return;
    int d = (t < E) ? dst[t] : (int)(t - E);
    float e = expf(logits[t] - f32_dec(menc[d]));
    logits[t] = e;                                   // reuse buffer to hold e
    atomicAdd(&denom[d], e);
}

// ---------- pass 3: weighted scatter aggregation, one wave per edge ----------
__global__ __launch_bounds__(256) void gat_edge_agg(const int* __restrict__ src,
                                                    const int* __restrict__ dst,
                                                    const float* __restrict__ h,
                                                    const float* __restrict__ logits,
                                                    const float* __restrict__ denom,
                                                    float* __restrict__ agg,
                                                    long long E, long long ET) {
    const int lane = threadIdx.x & 31;
    long long wid = ((long long)blockIdx.x * blockDim.x + threadIdx.x) >> 5;
    long long nw  = ((long long)gridDim.x * blockDim.x) >> 5;
    for (long long t = wid; t < ET; t += nw) {
        int s, d;
        if (t < E) { s = src[t]; d = dst[t]; }
        else       { s = d = (int)(t - E); }
        float alpha = logits[t] / denom[d];          // scalar broadcast loads
        float2 hv = *((const float2*)(h + (long long)s * HID) + lane);
        float* ap = agg + (long long)d * HID + lane * 2;
        atomicAdd(ap,     hv.x * alpha);
        atomicAdd(ap + 1, hv.y * alpha);
    }
}

// ---------- pass 4: sumvec[f] = sum_n relu(agg[n][f] + bias[f]) ----------
__global__ __launch_bounds__(256) void gat_reduce(const float* __restrict__ agg,
                                                  const float* __restrict__ bias,
                                                  float* __restrict__ sumvec,
                                                  long long N) {
    __shared__ float sdata[256];
    const int f    = threadIdx.x & 63;
    const int slot = threadIdx.x >> 6;               // 0..3
    const float bf = bias[f];
    float local = 0.0f;
    long long stride = (long long)gridDim.x * 4;
    for (long long n = (long long)blockIdx.x * 4 + slot; n < N; n += stride) {
        float v = agg[n * HID + f] + bf;
        local += (v > 0.0f) ? v : 0.0f;
    }
    sdata[threadIdx.x] = local;
    __syncthreads();
    if (threadIdx.x < 128) sdata[threadIdx.x] += sdata[threadIdx.x + 128];
    __syncthreads();
    if (threadIdx.x < 64) {
        float v = sdata[threadIdx.x] + sdata[threadIdx.x + 64];
        atomicAdd(&sumvec[threadIdx.x], v);
    }
}

// ---------- pass 5: out = (sumvec / N) @ W_lin + b_lin  (single block) ----------
__global__ void gat_final(const float* __restrict__ sumvec,
                          const float* __restrict__ W_lin,
                          const float* __restrict__ b_lin,
                          float* __restrict__ out, long long N) {
    int o = threadIdx.x;
    if (o >= HID) return;
    float invN = 1.0f / (float)N;
    float acc = b_lin[o];
    for (int f = 0; f < HID; ++f)
        acc += (sumvec[f] * invN) * W_lin[f * HID + o];
    out[o] = acc;
}

static inline size_t alignup(size_t v) { return (v + 255) & ~(size_t)255; }

extern "C" void kernel_launch(void* const* d_in, const int* in_sizes, int n_in,
                              void* d_out, int out_size, void* d_ws, size_t ws_size,
                              hipStream_t stream) {
    const float* x       = (const float*)d_in[0];
    const int*   ei      = (const int*)d_in[1];
    const float* W       = (const float*)d_in[2];
    const float* att_src = (const float*)d_in[3];
    const float* att_dst = (const float*)d_in[4];
    const float* bias    = (const float*)d_in[5];
    const float* W_lin   = (const float*)d_in[6];
    const float* b_lin   = (const float*)d_in[7];
    float* out = (float*)d_out;

    const long long N  = in_sizes[0] / 128;
    const long long E  = in_sizes[1] / 2;
    const long long ET = E + N;
    const int* src = ei;
    const int* dst = ei + E;

    // workspace layout
    char* w = (char*)d_ws;
    float*        h      = (float*)w;        w += alignup((size_t)N * HID * 4);
    float*        agg    = (float*)w;        w += alignup((size_t)N * HID * 4);
    float*        a_src  = (float*)w;        w += alignup((size_t)N * 4);
    float*        a_dst  = (float*)w;        w += alignup((size_t)N * 4);
    unsigned int* menc   = (unsigned int*)w; w += alignup((size_t)N * 4);
    float*        denom  = (float*)w;        w += alignup((size_t)N * 4);
    float*        logits = (float*)w;        w += alignup((size_t)ET * 4);
    float*        sumvec = (float*)w;        w += alignup((size_t)HID * 4);

    const int TB = 256;
    // init accumulators
    {
        long long tot = N * HID;
        dim3 g((unsigned)((tot + TB - 1) / TB));
        gat_init<<<g, TB, 0, stream>>>(agg, menc, denom, sumvec, N);
    }
    // h = x @ W  (WMMA f32 16x16x4)
    {
        dim3 g((unsigned)((N + 31) / 32));
        gat_gemm_h<<<g, TB, 0, stream>>>(x, W, h, N);
    }
    // attention scores
    {
        dim3 g((unsigned)((N + TB - 1) / TB));
        gat_scores<<<g, TB, 0, stream>>>(h, att_src, att_dst, a_src, a_dst, N);
    }
    // segment softmax: max, exp+sum
    {
        dim3 g((unsigned)((ET + TB - 1) / TB));
        gat_edge_max<<<g, TB, 0, stream>>>(src, dst, a_src, a_dst, logits, menc, E, ET);
        gat_edge_expsum<<<g, TB, 0, stream>>>(dst, logits, menc, denom, E, ET);
    }
    // weighted aggregation (one wave per edge, L2-resident atomics)
    {
        dim3 g(4096);
        gat_edge_agg<<<g, TB, 0, stream>>>(src, dst, h, logits, denom, agg, E, ET);
    }
    // relu + mean-commuted reduction, then tiny GEMV epilogue
    {
        dim3 g(1024);
        gat_reduce<<<g, TB, 0, stream>>>(agg, bias, sumvec, N);
        gat_final<<<1, 64, 0, stream>>>(sumvec, W_lin, b_lin, out, N);
    }
    (void)n_in; (void)out_size; (void)ws_size;
}